// MyGINConv_72086731096479
// MI455X (gfx1250) — compile-verified
//
#include <hip/hip_runtime.h>

#define N_NODES 100000
#define N_EDGES 1600000
#define D       128
#define NEG_SLOPE 0.01f

typedef __attribute__((ext_vector_type(2))) float v2f;
typedef __attribute__((ext_vector_type(8))) float v8f;

// ---------------------------------------------------------------------------
// Kernel 1: h = x   (eps = 0, so (1+eps)*x == x; scatter adds on top of this)
// ---------------------------------------------------------------------------
__global__ void gin_init_h(const float* __restrict__ x, float* __restrict__ h) {
    size_t i = (size_t)blockIdx.x * blockDim.x + threadIdx.x;   // one float4 each
    const float4* xs = (const float4*)x;
    float4*       hs = (float4*)h;
    hs[i] = xs[i];
}

// ---------------------------------------------------------------------------
// Kernel 2: h[dst] += x[src] over all edges.
// One wave32 per edge: lane l handles features [4l, 4l+4). x is L2-resident
// (51 MB < 192 MB L2), so gathers and atomics stay in L2.
// edge_index is int64, shape (2, N_EDGES): row 0 = src, row 1 = dst.
// ---------------------------------------------------------------------------
__global__ void gin_scatter_add(const float* __restrict__ x,
                                const long long* __restrict__ ei,
                                float* __restrict__ h) {
    size_t gid  = (size_t)blockIdx.x * blockDim.x + threadIdx.x;
    size_t edge = gid >> 5;
    int    lane = threadIdx.x & 31;
    if (edge >= N_EDGES) return;

    long long s = ei[edge];                     // source node j
    long long d = ei[(size_t)N_EDGES + edge];   // target node i

    const float4 v = *(const float4*)(x + (size_t)s * D + lane * 4);
    float* hp = h + (size_t)d * D + lane * 4;
    atomicAdd(hp + 0, v.x);
    atomicAdd(hp + 1, v.y);
    atomicAdd(hp + 2, v.z);
    atomicAdd(hp + 3, v.w);
}

// ---------------------------------------------------------------------------
// Kernel 3: Out = leaky_relu(A @ W + bias), A: [N_NODES x 128], W: [128 x 128].
// Block = 256 threads = 8 waves. Block owns 16 rows; wave w owns the 16-column
// tile starting at 16*w. K accumulated in steps of 4 with V_WMMA_F32_16X16X4_F32
// (exact f32 math). A tile staged once in LDS and shared by all 8 waves.
//
// Fragment layouts (ISA 7.12.2):
//   A 16x4 f32, 2 VGPRs: lanes 0-15 -> M=lane, {K=0,K=1}; lanes 16-31 -> {K=2,K=3}
//   B 4x16 f32, 2 VGPRs: lanes 0-15 -> N=lane, {K=0,K=1}; lanes 16-31 -> {K=2,K=3}
//   C/D 16x16 f32, 8 VGPRs: VGPR v -> M=v (lanes 0-15) / M=v+8 (lanes 16-31), N=lane%16
// Safe to run in-place (Out == A): all global reads of the block's rows happen
// during LDS staging, before any write, and no other block touches those rows.
// ---------------------------------------------------------------------------
__global__ void gin_gemm_bias_lrelu(const float* A,
                                    const float* __restrict__ W,
                                    const float* __restrict__ bias,
                                    float* Out) {
    __shared__ float As[16 * D];                 // 8 KB A tile

    const int tid  = threadIdx.x;
    const int lane = tid & 31;
    const int wave = tid >> 5;                   // 0..7
    const size_t row0 = (size_t)blockIdx.x * 16;

    // Stage A tile: 2048 floats = 512 float4s, 256 threads x 2
    {
        const float4* src = (const float4*)(A + row0 * D);
        float4*       dst = (float4*)As;
        dst[tid]       = src[tid];
        dst[tid + 256] = src[tid + 256];
    }
    __syncthreads();

    const int col0  = wave * 16;
    const int mn    = lane & 15;                 // M for A frag, N for B frag
    const int khalf = (lane >> 4) * 2;           // 0 (lanes 0-15) or 2 (lanes 16-31)

    v8f acc = {};
    #pragma unroll 4
    for (int k = 0; k < D; k += 4) {
        // A fragment: two consecutive K values for row mn -> ds_load_b64
        v2f a = *(const v2f*)(As + mn * D + k + khalf);
        // B fragment: column (col0+mn) of W at rows k+khalf, k+khalf+1
        v2f b;
        b.x = W[(size_t)(k + khalf)     * D + col0 + mn];
        b.y = W[(size_t)(k + khalf + 1) * D + col0 + mn];
        acc = __builtin_amdgcn_wmma_f32_16x16x4_f32(
                  /*neg_a=*/false, a, /*neg_b=*/false, b,
                  /*c_mod=*/(short)0, acc, /*reuse_a=*/false, /*reuse_b=*/false);
    }

    // Epilogue: bias + LeakyReLU, write D fragment
    const int   mbase = (lane >> 4) * 8;
    const float bn    = bias[col0 + mn];
    #pragma unroll
    for (int v = 0; v < 8; ++v) {
        float val = acc[v] + bn;
        val = (val >= 0.0f) ? val : NEG_SLOPE * val;
        Out[(row0 + mbase + v) * (size_t)D + col0 + mn] = val;
    }
}

// ---------------------------------------------------------------------------
extern "C" void kernel_launch(void* const* d_in, const int* in_sizes, int n_in,
                              void* d_out, int out_size, void* d_ws, size_t ws_size,
                              hipStream_t stream) {
    const float*     x  = (const float*)d_in[0];
    const long long* ei = (const long long*)d_in[1];   // int64 edge_index
    const float*     W1 = (const float*)d_in[2];
    const float*     b1 = (const float*)d_in[3];
    const float*     W2 = (const float*)d_in[4];
    const float*     b2 = (const float*)d_in[5];
    float*           out = (float*)d_out;
    float*           h   = (float*)d_ws;               // 100000*128 f32 = 51.2 MB

    // h = x
    gin_init_h<<<(N_NODES * D / 4) / 256, 256, 0, stream>>>(x, h);
    // h += scatter-add of x[src] into rows dst   (8 edges / 256-thread block)
    gin_scatter_add<<<N_EDGES / 8, 256, 0, stream>>>(x, ei, h);
    // h = leaky_relu(h @ W1 + b1)   (in-place: block-local rows only)
    gin_gemm_bias_lrelu<<<N_NODES / 16, 256, 0, stream>>>(h, W1, b1, h);
    // out = leaky_relu(h @ W2 + b2)
    gin_gemm_bias_lrelu<<<N_NODES / 16, 256, 0, stream>>>(h, W2, b2, out);
}